// PointNet2Segmenter_v1_15324443312200
// MI455X (gfx1250) — compile-verified
//
#include <hip/hip_runtime.h>
#include <cstdint>
#include <cstddef>

#define BATCH 16
#define NPTS  4096

typedef __attribute__((ext_vector_type(2))) float f32x2;
typedef __attribute__((ext_vector_type(8))) float f32x8;

// ------------------------------------------------------------------
// transpose input [B,3,N] -> xyz [B,N,3]
// ------------------------------------------------------------------
__global__ void k_transpose(const float* __restrict__ in, float* __restrict__ xyz)
{
    int tid = blockIdx.x * blockDim.x + threadIdx.x;
    if (tid >= BATCH * NPTS) return;
    int b = tid / NPTS, n = tid % NPTS;
    const float* src = in + (size_t)b * 3 * NPTS + n;
    float* dst = xyz + (size_t)tid * 3;
    dst[0] = src[0];
    dst[1] = src[NPTS];
    dst[2] = src[2 * NPTS];
}

// ------------------------------------------------------------------
// farthest point sampling: one block per batch, dist kept in LDS
// ------------------------------------------------------------------
__global__ __launch_bounds__(256) void k_fps(const float* __restrict__ xyz,
                                             int Npts, int S,
                                             float* __restrict__ out_xyz)
{
    __shared__ float dist[4096];
    __shared__ float rv[256];
    __shared__ int   ri[256];
    int b = blockIdx.x, t = threadIdx.x;
    const float* P = xyz + (size_t)b * Npts * 3;
    for (int n = t; n < Npts; n += 256) dist[n] = 1e10f;
    __syncthreads();

    int far = 0;
    for (int s = 0; s < S; ++s) {
        float cx = P[far * 3 + 0], cy = P[far * 3 + 1], cz = P[far * 3 + 2];
        if (t == 0) {
            float* o = out_xyz + ((size_t)b * S + s) * 3;
            o[0] = cx; o[1] = cy; o[2] = cz;
        }
        float bv = -1.0f; int bi = 0x7fffffff;
        for (int n = t; n < Npts; n += 256) {
            float dx = P[n * 3 + 0] - cx;
            float dy = P[n * 3 + 1] - cy;
            float dz = P[n * 3 + 2] - cz;
            float d  = dx * dx + dy * dy + dz * dz;
            float dm = dist[n];
            dm = d < dm ? d : dm;
            dist[n] = dm;
            if (dm > bv) { bv = dm; bi = n; }   // ascending n => first-max kept
        }
        rv[t] = bv; ri[t] = bi;
        __syncthreads();
        for (int off = 128; off > 0; off >>= 1) {
            if (t < off) {
                float ov = rv[t + off]; int oi = ri[t + off];
                if (ov > rv[t] || (ov == rv[t] && oi < ri[t])) { rv[t] = ov; ri[t] = oi; }
            }
            __syncthreads();
        }
        far = ri[0];
        __syncthreads();
    }
}

// ------------------------------------------------------------------
// ball query: one wave32 per (b,s); ballot + prefix-popcount emits the
// lowest-index hits in ascending order (== jnp.sort of masked indices)
// ------------------------------------------------------------------
__global__ __launch_bounds__(128) void k_ballquery(const float* __restrict__ xyz, int Npts,
                                                   const float* __restrict__ nq, int S,
                                                   float r2, int K,
                                                   int* __restrict__ idx)
{
    __shared__ int sidx[4][64];
    int lane = threadIdx.x & 31;
    int w    = threadIdx.x >> 5;
    int gs   = blockIdx.x * 4 + w;          // (b*S + s), grid launched exactly
    int b = gs / S;
    const float* q = nq + (size_t)gs * 3;
    float qx = q[0], qy = q[1], qz = q[2];
    const float* P = xyz + (size_t)b * Npts * 3;

    int cnt = 0;
    for (int base = 0; base < Npts && cnt < K; base += 32) {
        int n = base + lane;                 // Npts is a multiple of 32
        float dx = P[n * 3 + 0] - qx;
        float dy = P[n * 3 + 1] - qy;
        float dz = P[n * 3 + 2] - qz;
        float d2 = dx * dx + dy * dy + dz * dz;
        bool hit = (d2 <= r2);
        unsigned mask = (unsigned)__ballot(hit);
        int pos = cnt + __popc(mask & ((1u << lane) - 1u));
        if (hit && pos < K) sidx[w][pos] = n;
        cnt += __popc(mask);
    }
    if (cnt > K) cnt = K;
    __syncthreads();
    int first = sidx[w][0];                  // query center always hits itself
    int* oidx = idx + (size_t)gs * K;
    for (int k = lane; k < K; k += 32) oidx[k] = (k < cnt) ? sidx[w][k] : first;
}

// ------------------------------------------------------------------
// grouping: act[c, m] with m=(b*S+s)*K+k
// ------------------------------------------------------------------
__global__ void k_group(const float* __restrict__ xyz, int Npts,
                        const float* __restrict__ nq, int S, int K,
                        const int* __restrict__ idx,
                        const float* __restrict__ feats, int Cf,
                        float* __restrict__ act, int M)
{
    int tid = blockIdx.x * blockDim.x + threadIdx.x;
    if (tid >= M) return;
    int b = tid / (S * K);
    int r = tid % (S * K);
    int s = r / K;
    int i = idx[tid];
    const float* p = xyz + ((size_t)b * Npts + i) * 3;
    const float* q = nq + ((size_t)b * S + s) * 3;
    act[tid]                 = p[0] - q[0];
    act[(size_t)M + tid]     = p[1] - q[1];
    act[(size_t)2 * M + tid] = p[2] - q[2];
    for (int c = 0; c < Cf; ++c)
        act[(size_t)(3 + c) * M + tid] = feats[((size_t)b * Cf + c) * Npts + i];
}

// group-all for SA3: ch 0..2 = l2_xyz (absolute), ch 3..258 = l2_points
__global__ void k_group_all(const float* __restrict__ l2xyz,
                            const float* __restrict__ l2p,
                            float* __restrict__ act)
{
    const int M = BATCH * 128;
    int tid = blockIdx.x * blockDim.x + threadIdx.x;
    if (tid >= M) return;
    int b = tid / 128, p = tid % 128;
    const float* q = l2xyz + (size_t)tid * 3;
    act[tid]                 = q[0];
    act[(size_t)M + tid]     = q[1];
    act[(size_t)2 * M + tid] = q[2];
    for (int c = 0; c < 256; ++c)
        act[(size_t)(3 + c) * M + tid] = l2p[((size_t)b * 256 + c) * 128 + p];
}

// ------------------------------------------------------------------
// f32 WMMA GEMM (main): out[o,m] = bias[o] + sum_c W[o*Cin+c]*A[c*M+m]
// Requires: ldW == Cin, M % 64 == 0, O % 16 == 0, Cin <= 512.
// One wave -> 16(o) x 64(m); weight tile staged block-wide into LDS
// via async global->LDS copies; K unrolled by 8, branch-free body.
// ------------------------------------------------------------------
__global__ __launch_bounds__(256) void k_gemm_wmma(const float* __restrict__ W,
                                                   int Cin, int O,
                                                   const float* __restrict__ A, int M,
                                                   const float* __restrict__ bias,
                                                   float* __restrict__ out)
{
    __shared__ __align__(16) float wlds[16 * 520];      // 16 x (Cin padded), <= 33 KB
    const int stride = Cin + ((Cin & 1) ? 1 : 2);       // even + bank-skewed

    const int lane  = threadIdx.x & 31;
    const int wave  = threadIdx.x >> 5;
    const int hf    = lane >> 4;
    const int l     = lane & 15;
    const int o0    = blockIdx.y << 4;
    const int mbase = (blockIdx.x * 8 + wave) << 6;

    // ---- stage 16 x Cin weight tile into LDS (async global->LDS) ----
    {
        const float* Wt = W + (size_t)o0 * Cin;
        unsigned lbase = (unsigned)(size_t)&wlds[0];
        for (int i = threadIdx.x; i < 16 * Cin; i += 256) {
            int r = i / Cin, c = i - r * Cin;
            unsigned loff = lbase + (unsigned)(r * stride + c) * 4u;
            unsigned long long ga = (unsigned long long)(const void*)(Wt + i);
            asm volatile("global_load_async_to_lds_b32 %0, %1, off"
                         :: "v"(loff), "v"(ga) : "memory");
        }
        asm volatile("s_wait_asynccnt 0x0" ::: "memory");
    }
    __syncthreads();
    if (mbase >= M) return;                 // wave-uniform

    f32x8 acc0 = {}, acc1 = {}, acc2 = {}, acc3 = {};
    const int kofs = 2 * hf;                // lane-half K pair
    const int kmain = Cin & ~7;
    const size_t step = (size_t)4 * M;

    const float* pa = A + (size_t)kofs * M + mbase + l;   // row kofs
    const float* pb = pa + (size_t)M;                     // row kofs+1

    for (int k = 0; k < kmain; k += 8) {
        if (k + 8 < kmain) __builtin_prefetch(pa + 2 * step, 0, 1);   // uniform guard
        f32x2 a0 = *(const f32x2*)&wlds[l * stride + k + kofs];
        f32x2 a1 = *(const f32x2*)&wlds[l * stride + k + 4 + kofs];
        const float* qa = pa + step;
        const float* qb = pb + step;
        f32x2 b00 = { pa[0],  pb[0]  };
        f32x2 b01 = { pa[16], pb[16] };
        f32x2 b02 = { pa[32], pb[32] };
        f32x2 b03 = { pa[48], pb[48] };
        f32x2 b10 = { qa[0],  qb[0]  };
        f32x2 b11 = { qa[16], qb[16] };
        f32x2 b12 = { qa[32], qb[32] };
        f32x2 b13 = { qa[48], qb[48] };
        acc0 = __builtin_amdgcn_wmma_f32_16x16x4_f32(false, a0, false, b00, (short)0, acc0, false, false);
        acc1 = __builtin_amdgcn_wmma_f32_16x16x4_f32(false, a0, false, b01, (short)0, acc1, false, false);
        acc2 = __builtin_amdgcn_wmma_f32_16x16x4_f32(false, a0, false, b02, (short)0, acc2, false, false);
        acc3 = __builtin_amdgcn_wmma_f32_16x16x4_f32(false, a0, false, b03, (short)0, acc3, false, false);
        acc0 = __builtin_amdgcn_wmma_f32_16x16x4_f32(false, a1, false, b10, (short)0, acc0, false, false);
        acc1 = __builtin_amdgcn_wmma_f32_16x16x4_f32(false, a1, false, b11, (short)0, acc1, false, false);
        acc2 = __builtin_amdgcn_wmma_f32_16x16x4_f32(false, a1, false, b12, (short)0, acc2, false, false);
        acc3 = __builtin_amdgcn_wmma_f32_16x16x4_f32(false, a1, false, b13, (short)0, acc3, false, false);
        pa += 2 * step;
        pb += 2 * step;
    }

    // ---- K tail (Cin % 8 in {0,3}): guarded, at most one k-step pair ----
    for (int k = kmain; k < Cin; k += 4) {
        int ca = k + kofs, cb = ca + 1;
        float ax = (ca < Cin) ? wlds[l * stride + ca] : 0.0f;
        float ay = (cb < Cin) ? wlds[l * stride + cb] : 0.0f;
        f32x2 a = {ax, ay};
        const float* ra = A + (size_t)((ca < Cin) ? ca : 0) * M + mbase + l;
        const float* rb = A + (size_t)((cb < Cin) ? cb : 0) * M + mbase + l;
        f32x2 b0 = { ra[0],  rb[0]  };
        f32x2 b1 = { ra[16], rb[16] };
        f32x2 b2 = { ra[32], rb[32] };
        f32x2 b3 = { ra[48], rb[48] };
        acc0 = __builtin_amdgcn_wmma_f32_16x16x4_f32(false, a, false, b0, (short)0, acc0, false, false);
        acc1 = __builtin_amdgcn_wmma_f32_16x16x4_f32(false, a, false, b1, (short)0, acc1, false, false);
        acc2 = __builtin_amdgcn_wmma_f32_16x16x4_f32(false, a, false, b2, (short)0, acc2, false, false);
        acc3 = __builtin_amdgcn_wmma_f32_16x16x4_f32(false, a, false, b3, (short)0, acc3, false, false);
    }

#pragma unroll
    for (int r = 0; r < 8; ++r) {
        int o = o0 + r + 8 * hf;            // C/D layout: VGPR r, half -> M = r + 8*half
        float bv = bias[o];
        out[(size_t)o * M + mbase + l]      = acc0[r] + bv;
        out[(size_t)o * M + mbase + 16 + l] = acc1[r] + bv;
        out[(size_t)o * M + mbase + 32 + l] = acc2[r] + bv;
        out[(size_t)o * M + mbase + 48 + l] = acc3[r] + bv;
    }
}

// ------------------------------------------------------------------
// small WMMA GEMM for the head base term: M == 16 (one 16x16 tile per
// wave), arbitrary ldW (1027), Cin % 4 == 0. out[o*16 + b].
// ------------------------------------------------------------------
__global__ __launch_bounds__(256) void k_gemm_small(const float* __restrict__ W, int ldW,
                                                    int Cin, int O,
                                                    const float* __restrict__ A,
                                                    const float* __restrict__ bias,
                                                    float* __restrict__ out)
{
    const int lane = threadIdx.x & 31;
    const int wave = threadIdx.x >> 5;
    const int hf   = lane >> 4;
    const int l    = lane & 15;
    const int o0   = (blockIdx.x * 8 + wave) << 4;
    if (o0 >= O) return;

    f32x8 acc = {};
    const int kofs = 2 * hf;
    const float* wr = W + (size_t)(o0 + l) * ldW + kofs;
    const float* ar = A + (size_t)kofs * 16 + l;
    for (int k = 0; k < Cin; k += 4) {
        f32x2 a = { wr[0], wr[1] };
        f32x2 b = { ar[0], ar[16] };
        acc = __builtin_amdgcn_wmma_f32_16x16x4_f32(false, a, false, b, (short)0, acc, false, false);
        wr += 4;
        ar += 64;
    }
#pragma unroll
    for (int r = 0; r < 8; ++r) {
        int o = o0 + r + 8 * hf;
        out[(size_t)o * 16 + l] = acc[r] + bias[o];
    }
}

// ------------------------------------------------------------------
// BatchNorm (training mode) stats + in-place normalize/affine/ReLU
// ------------------------------------------------------------------
__global__ __launch_bounds__(256) void k_bn_stats(const float* __restrict__ A, int M,
                                                  float* __restrict__ mean,
                                                  float* __restrict__ rstd)
{
    __shared__ float s1[256], s2[256];
    int c = blockIdx.x, t = threadIdx.x;
    const float* row = A + (size_t)c * M;
    float a = 0.0f, b = 0.0f;
    for (int i = t; i < M; i += 256) { float x = row[i]; a += x; b += x * x; }
    s1[t] = a; s2[t] = b;
    __syncthreads();
    for (int off = 128; off > 0; off >>= 1) {
        if (t < off) { s1[t] += s1[t + off]; s2[t] += s2[t + off]; }
        __syncthreads();
    }
    if (t == 0) {
        float m = s1[0] / (float)M;
        float v = s2[0] / (float)M - m * m;
        mean[c] = m;
        rstd[c] = rsqrtf(v + 1e-5f);
    }
}

__global__ void k_bn_relu(float* __restrict__ A, int C, int M,
                          const float* __restrict__ mean, const float* __restrict__ rstd,
                          const float* __restrict__ g, const float* __restrict__ bt)
{
    size_t total = (size_t)C * M;
    for (size_t i = (size_t)blockIdx.x * blockDim.x + threadIdx.x; i < total;
         i += (size_t)gridDim.x * blockDim.x) {
        int c = (int)(i / (size_t)M);
        float x = A[i];
        x = g[c] * (x - mean[c]) * rstd[c] + bt[c];
        A[i] = x > 0.0f ? x : 0.0f;
    }
}

// ------------------------------------------------------------------
// max-pool over K
// ------------------------------------------------------------------
__global__ void k_maxpool(const float* __restrict__ act, int C, int S, int K, int M,
                          float* __restrict__ out)
{
    int tid = blockIdx.x * blockDim.x + threadIdx.x;
    if (tid >= BATCH * C * S) return;
    int b = tid / (C * S);
    int r = tid % (C * S);
    int c = r / S, s = r % S;
    const float* p = act + (size_t)c * M + ((size_t)b * S + s) * K;
    float m = p[0];
    for (int k = 1; k < K; ++k) m = p[k] > m ? p[k] : m;
    out[((size_t)b * C + c) * S + s] = m;
}

// max over all 128 points -> l3t stored transposed [c*16 + b]
__global__ void k_maxpool_all(const float* __restrict__ act, int C, int P,
                              float* __restrict__ l3t)
{
    int tid = blockIdx.x * blockDim.x + threadIdx.x;
    if (tid >= C * BATCH) return;
    int c = tid / BATCH, b = tid % BATCH;
    const float* p = act + (size_t)c * (BATCH * P) + (size_t)b * P;
    float m = p[0];
    for (int k = 1; k < P; ++k) m = p[k] > m ? p[k] : m;
    l3t[(size_t)c * BATCH + b] = m;
}

// ------------------------------------------------------------------
// head layer-0 fusion: h0[o,m] = base[o,b] + W0[o,1024..1026]·xyz(b,n)
// ------------------------------------------------------------------
__global__ void k_head0(const float* __restrict__ base0, const float* __restrict__ W0,
                        const float* __restrict__ in, float* __restrict__ out)
{
    const size_t total = (size_t)512 * 65536;
    for (size_t i = (size_t)blockIdx.x * blockDim.x + threadIdx.x; i < total;
         i += (size_t)gridDim.x * blockDim.x) {
        int o = (int)(i >> 16);
        int m = (int)(i & 65535);
        int b = m >> 12;
        int n = m & 4095;
        const float* xin = in + (size_t)b * 3 * NPTS;
        const float* wr  = W0 + (size_t)o * 1027;
        out[i] = base0[o * BATCH + b]
               + wr[1024] * xin[n]
               + wr[1025] * xin[NPTS + n]
               + wr[1026] * xin[2 * NPTS + n];
    }
}

// final 1x1 conv (O=2, no BN) + transpose to [B,N,2]
__global__ void k_conv4(const float* __restrict__ h, const float* __restrict__ W4,
                        const float* __restrict__ b4, float* __restrict__ out)
{
    const int M = BATCH * NPTS;
    int m = blockIdx.x * blockDim.x + threadIdx.x;
    if (m >= M) return;
    float a0 = b4[0], a1 = b4[1];
    for (int c = 0; c < 128; ++c) {
        float x = h[(size_t)c * M + m];
        a0 += W4[c] * x;
        a1 += W4[128 + c] * x;
    }
    out[(size_t)m * 2 + 0] = a0;
    out[(size_t)m * 2 + 1] = a1;
}

// ==================================================================
extern "C" void kernel_launch(void* const* d_in, const int* in_sizes, int n_in,
                              void* d_out, int out_size, void* d_ws, size_t ws_size,
                              hipStream_t stream)
{
    const float* input = (const float*)d_in[0];

    // Resolve pytree flattening order: sorted dict keys (conv4 first, size 256)
    // vs insertion order (sa1 W0 first, size 192).
    int i_sa1, i_sa2, i_sa3, i_head, i_conv4;
    if (in_sizes[1] == 256) { i_conv4 = 1; i_head = 3; i_sa1 = 15; i_sa2 = 27; i_sa3 = 39; }
    else                    { i_sa1 = 1; i_sa2 = 13; i_sa3 = 25; i_head = 37; i_conv4 = 49; }
    auto P = [&](int i) { return (const float*)d_in[i]; };

    // ---- workspace bump allocator
    char* wsb = (char*)d_ws;
    size_t off = 0;
    auto alloc = [&](size_t nf) {
        void* p = wsb + off;
        off = (off + nf * sizeof(float) + 255) & ~(size_t)255;
        return p;
    };
    float* xyz   = (float*)alloc((size_t)BATCH * NPTS * 3);
    float* nx1   = (float*)alloc((size_t)BATCH * 512 * 3);
    int*   idx1  = (int*)  alloc((size_t)BATCH * 512 * 32);
    float* l1p   = (float*)alloc((size_t)BATCH * 128 * 512);
    float* nx2   = (float*)alloc((size_t)BATCH * 128 * 3);
    int*   idx2  = (int*)  alloc((size_t)BATCH * 128 * 64);
    float* l2p   = (float*)alloc((size_t)BATCH * 256 * 128);
    float* l3t   = (float*)alloc((size_t)1024 * BATCH);
    float* base0 = (float*)alloc((size_t)512 * BATCH);
    float* meanA = (float*)alloc(1024);
    float* rstdA = (float*)alloc(1024);
    float* bufA  = (float*)alloc((size_t)33554432);
    float* bufB  = (float*)alloc((size_t)33554432);

    const int M1 = BATCH * 512 * 32;   // 262144
    const int M2 = BATCH * 128 * 64;   // 131072
    const int M3 = BATCH * 128;        // 2048
    const int MH = BATCH * NPTS;       // 65536

    auto gemm = [&](const float* W, int Cin, int O,
                    const float* A, int M, const float* bias, float* out) {
        dim3 g(M / 512, O / 16);       // all main-GEMM M are multiples of 512
        k_gemm_wmma<<<g, 256, 0, stream>>>(W, Cin, O, A, M, bias, out);
    };
    auto bn = [&](float* A, int C, int M, const float* g, const float* bt) {
        k_bn_stats<<<C, 256, 0, stream>>>(A, M, meanA, rstdA);
        k_bn_relu<<<4096, 256, 0, stream>>>(A, C, M, meanA, rstdA, g, bt);
    };
    auto LW  = [&](int base, int li) { return P(base + li * 4 + 0); };
    auto LB  = [&](int base, int li) { return P(base + li * 4 + 1); };
    auto LG  = [&](int base, int li) { return P(base + li * 4 + 2); };
    auto LBT = [&](int base, int li) { return P(base + li * 4 + 3); };

    // ---- SA1 ----------------------------------------------------
    k_transpose<<<(BATCH * NPTS + 255) / 256, 256, 0, stream>>>(input, xyz);
    k_fps<<<BATCH, 256, 0, stream>>>(xyz, NPTS, 512, nx1);
    k_ballquery<<<(BATCH * 512) / 4, 128, 0, stream>>>(xyz, NPTS, nx1, 512, 0.04f, 32, idx1);
    k_group<<<(M1 + 255) / 256, 256, 0, stream>>>(xyz, NPTS, nx1, 512, 32, idx1, nullptr, 0, bufA, M1);
    gemm(LW(i_sa1, 0), 3, 64, bufA, M1, LB(i_sa1, 0), bufB);    bn(bufB, 64, M1, LG(i_sa1, 0), LBT(i_sa1, 0));
    gemm(LW(i_sa1, 1), 64, 64, bufB, M1, LB(i_sa1, 1), bufA);   bn(bufA, 64, M1, LG(i_sa1, 1), LBT(i_sa1, 1));
    gemm(LW(i_sa1, 2), 64, 128, bufA, M1, LB(i_sa1, 2), bufB);  bn(bufB, 128, M1, LG(i_sa1, 2), LBT(i_sa1, 2));
    k_maxpool<<<(BATCH * 128 * 512 + 255) / 256, 256, 0, stream>>>(bufB, 128, 512, 32, M1, l1p);

    // ---- SA2 ----------------------------------------------------
    k_fps<<<BATCH, 256, 0, stream>>>(nx1, 512, 128, nx2);
    k_ballquery<<<(BATCH * 128) / 4, 128, 0, stream>>>(nx1, 512, nx2, 128, 0.16f, 64, idx2);
    k_group<<<(M2 + 255) / 256, 256, 0, stream>>>(nx1, 512, nx2, 128, 64, idx2, l1p, 128, bufA, M2);
    gemm(LW(i_sa2, 0), 131, 128, bufA, M2, LB(i_sa2, 0), bufB); bn(bufB, 128, M2, LG(i_sa2, 0), LBT(i_sa2, 0));
    gemm(LW(i_sa2, 1), 128, 128, bufB, M2, LB(i_sa2, 1), bufA); bn(bufA, 128, M2, LG(i_sa2, 1), LBT(i_sa2, 1));
    gemm(LW(i_sa2, 2), 128, 256, bufA, M2, LB(i_sa2, 2), bufB); bn(bufB, 256, M2, LG(i_sa2, 2), LBT(i_sa2, 2));
    k_maxpool<<<(BATCH * 256 * 128 + 255) / 256, 256, 0, stream>>>(bufB, 256, 128, 64, M2, l2p);

    // ---- SA3 (group all) ---------------------------------------
    k_group_all<<<(M3 + 255) / 256, 256, 0, stream>>>(nx2, l2p, bufA);
    gemm(LW(i_sa3, 0), 259, 256, bufA, M3, LB(i_sa3, 0), bufB);  bn(bufB, 256, M3, LG(i_sa3, 0), LBT(i_sa3, 0));
    gemm(LW(i_sa3, 1), 256, 512, bufB, M3, LB(i_sa3, 1), bufA);  bn(bufA, 512, M3, LG(i_sa3, 1), LBT(i_sa3, 1));
    gemm(LW(i_sa3, 2), 512, 1024, bufA, M3, LB(i_sa3, 2), bufB); bn(bufB, 1024, M3, LG(i_sa3, 2), LBT(i_sa3, 2));
    k_maxpool_all<<<(1024 * BATCH + 255) / 256, 256, 0, stream>>>(bufB, 1024, 128, l3t);

    // ---- head --------------------------------------------------
    // base[o,b] = b0[o] + W0[o, :1024] @ l3[:, b]   (ldW = 1027)
    k_gemm_small<<<4, 256, 0, stream>>>(LW(i_head, 0), 1027, 1024, 512, l3t, LB(i_head, 0), base0);
    k_head0<<<8192, 256, 0, stream>>>(base0, LW(i_head, 0), input, bufA);
    bn(bufA, 512, MH, LG(i_head, 0), LBT(i_head, 0));
    gemm(LW(i_head, 1), 512, 256, bufA, MH, LB(i_head, 1), bufB); bn(bufB, 256, MH, LG(i_head, 1), LBT(i_head, 1));
    gemm(LW(i_head, 2), 256, 128, bufB, MH, LB(i_head, 2), bufA); bn(bufA, 128, MH, LG(i_head, 2), LBT(i_head, 2));

    // ---- conv4 + transpose to [B,N,2] --------------------------
    k_conv4<<<(MH + 255) / 256, 256, 0, stream>>>(bufA, P(i_conv4), P(i_conv4 + 1), (float*)d_out);
}